// RGCNLinkPredictor_43954695308082
// MI455X (gfx1250) — compile-verified
//
#include <hip/hip_runtime.h>

// ---------------------------------------------------------------------------
// RGCN link predictor for gfx1250 (MI455X), fp32 end-to-end.
//   layer: mean-aggregate per (dst, rel) into agg[N, R*D], then
//          out = (agg * inv_cnt) @ W[R*D, D] + x @ root + b   (WMMA f32 16x16x4)
//   score: per 16-query tile, T_r = ZH @ rel_W[r] for all r, rowdot with ZT,
//          select by rel_ids (mirrors the reference's stack-then-select).
// Weights are pre-transposed (col-major) so each B fragment (rows ka, ka+1 at
// column n) is one contiguous 8-byte load instead of two strided b32 loads.
// ---------------------------------------------------------------------------

typedef __attribute__((ext_vector_type(2))) float v2f;
typedef __attribute__((ext_vector_type(8))) float v8f;

#define RGCN_D 128
#define RGCN_R 8
#define RGCN_KA (RGCN_R * RGCN_D)   // 1024
#define RGCN_MAT (RGCN_D * RGCN_D)  // 16384

__device__ __forceinline__ v8f wmma_f32_4(v2f a, v2f b, v8f c) {
  // 8 args: (neg_a, A, neg_b, B, c_mod, C, reuse_a, reuse_b)
  return __builtin_amdgcn_wmma_f32_16x16x4_f32(
      false, a, false, b, (short)0, c, false, false);
}

// ---- batched 128x128 transpose: out[m][n][d] = in[m][d][n] ----------------
__global__ void __launch_bounds__(256)
rgcn_transpose_kernel(const float* __restrict__ in, float* __restrict__ out,
                      int nmat) {
  int idx = blockIdx.x * blockDim.x + threadIdx.x;
  int total = nmat * RGCN_MAT;
  if (idx >= total) return;
  int m = idx >> 14;            // matrix
  int d = (idx >> 7) & 127;     // row in source
  int n = idx & 127;            // col in source (coalesced reads)
  out[(size_t)m * RGCN_MAT + (size_t)n * RGCN_D + d] = in[idx];
}

// ---- counts: cnt[dst*R + rel] += 1 ----------------------------------------
__global__ void __launch_bounds__(256)
rgcn_count_kernel(const int* __restrict__ dst, const int* __restrict__ et,
                  float* __restrict__ cnt, int E) {
  int e = blockIdx.x * blockDim.x + threadIdx.x;
  if (e < E) atomicAdd(&cnt[(size_t)dst[e] * RGCN_R + et[e]], 1.0f);
}

// ---- cnt -> 1/max(cnt,1) ---------------------------------------------------
__global__ void __launch_bounds__(256)
rgcn_inv_kernel(float* __restrict__ cnt, int n) {
  int i = blockIdx.x * blockDim.x + threadIdx.x;
  if (i < n) cnt[i] = 1.0f / fmaxf(cnt[i], 1.0f);
}

// ---- agg[(dst*R+rel), :] += x[src, :]  (float4 per thread, f32 atomics) ---
__global__ void __launch_bounds__(256)
rgcn_scatter_kernel(const int* __restrict__ src, const int* __restrict__ dst,
                    const int* __restrict__ et, const float* __restrict__ x,
                    float* __restrict__ agg, int E) {
  int idx = blockIdx.x * blockDim.x + threadIdx.x;
  int e = idx >> 5;                 // 32 threads per edge, 4 floats each
  if (e >= E) return;
  int j = (idx & 31) * 4;
  const float4 v = *(const float4*)(x + (size_t)src[e] * RGCN_D + j);
  float* p = agg + ((size_t)dst[e] * RGCN_R + et[e]) * RGCN_D + j;
  atomicAdd(p + 0, v.x);
  atomicAdd(p + 1, v.y);
  atomicAdd(p + 2, v.z);
  atomicAdd(p + 3, v.w);
}

// ---- layer GEMM: out[n,:] = sum_r (agg[n,r,:]*inv[n,r]) @ W_r + x@root + b -
// One wave = 16 rows x 128 cols (8 C tiles). 8 waves/block -> 128 rows/block.
// Wt / roott are TRANSPOSED (col-major): Wt[r][n][d] = W[r][d][n].
__global__ void __launch_bounds__(256)
rgcn_layer_kernel(const float* __restrict__ agg,    // [N, R*D]
                  const float* __restrict__ inv,    // [N, R]
                  const float* __restrict__ xin,    // [N, D]
                  const float* __restrict__ Wt,     // [R, D(n), D(d)] transposed
                  const float* __restrict__ roott,  // [D(n), D(d)] transposed
                  const float* __restrict__ bias,   // [D]
                  float* __restrict__ out,          // [N, D]
                  int n_nodes, int do_relu) {
  const int lane = threadIdx.x & 31;
  const int wave = threadIdx.x >> 5;
  const int lo = lane & 15;
  const int hi = lane >> 4;
  const int m0 = blockIdx.x * 128 + wave * 16;
  if (m0 >= n_nodes) return;        // wave-uniform: EXEC stays all-ones
  int row = m0 + lo;                // A-matrix row this lane loads
  if (row >= n_nodes) row = n_nodes - 1;   // clamp (stores masked later)

  v8f acc[8] = {};                  // 8 column tiles of 16

  const float* arow = agg + (size_t)row * RGCN_KA;
  const float* irow = inv + (size_t)row * RGCN_R;

  // Phase 1: relation einsum, K = R*D, scale A rows by inv-count per relation
  for (int r = 0; r < RGCN_R; ++r) {
    const float s = irow[r];
    const float* ar = arow + r * RGCN_D;
    const float* Wr = Wt + (size_t)r * RGCN_MAT;
    for (int k0 = 0; k0 < RGCN_D; k0 += 4) {
      const int ka = k0 + 2 * hi;   // A layout: lane holds k = 2*hi + {0,1}
      v2f a = *(const v2f*)(ar + ka);
      a *= s;
#pragma unroll
      for (int ct = 0; ct < 8; ++ct) {
        const int n = ct * 16 + lo; // B layout: lane holds rows 2*hi+{0,1}, col n
        const v2f b = *(const v2f*)(Wr + (size_t)n * RGCN_D + ka);
        acc[ct] = wmma_f32_4(a, b, acc[ct]);
      }
    }
  }

  // Phase 2: root term, K = D
  const float* xrow = xin + (size_t)row * RGCN_D;
  for (int k0 = 0; k0 < RGCN_D; k0 += 4) {
    const int ka = k0 + 2 * hi;
    const v2f a = *(const v2f*)(xrow + ka);
#pragma unroll
    for (int ct = 0; ct < 8; ++ct) {
      const int n = ct * 16 + lo;
      const v2f b = *(const v2f*)(roott + (size_t)n * RGCN_D + ka);
      acc[ct] = wmma_f32_4(a, b, acc[ct]);
    }
  }

  // Epilogue: C layout -> lane L, vgpr vc holds C[vc + 8*hi][lo]
#pragma unroll
  for (int ct = 0; ct < 8; ++ct) {
    const int n = ct * 16 + lo;
    const float bv = bias[n];
#pragma unroll
    for (int vc = 0; vc < 8; ++vc) {
      const int orow = m0 + vc + 8 * hi;
      if (orow < n_nodes) {
        float v = acc[ct][vc] + bv;
        if (do_relu) v = fmaxf(v, 0.0f);
        out[(size_t)orow * RGCN_D + n] = v;
      }
    }
  }
}

// ---- scoring: one wave per 16-query tile ----------------------------------
// relWt is TRANSPOSED: relWt[r][n][d] = rel_W[r][d][n].
__global__ void __launch_bounds__(256)
rgcn_score_kernel(const float* __restrict__ z,       // [N, D]
                  const int* __restrict__ head,      // [Q]
                  const int* __restrict__ tail,      // [Q]
                  const int* __restrict__ rel,       // [Q]
                  const float* __restrict__ relWt,   // [R, D(n), D(d)]
                  float* __restrict__ out, int Q) {
  const int lane = threadIdx.x & 31;
  const int wave = threadIdx.x >> 5;
  const int lo = lane & 15;
  const int hi = lane >> 4;
  const int q0 = (blockIdx.x * 8 + wave) * 16;
  if (q0 >= Q) return;              // wave-uniform

  // A rows: z[head[q0+lo]]  (lanes 0-15 and 16-31 both cover m = 0..15)
  int qa = q0 + lo;
  if (qa >= Q) qa = Q - 1;
  const float* zh = z + (size_t)head[qa] * RGCN_D;

  // Per-lane metadata for the C-rows this lane owns: m = vc + 8*hi
  int trow[8], qr[8];
#pragma unroll
  for (int vc = 0; vc < 8; ++vc) {
    int qm = q0 + vc + 8 * hi;
    if (qm >= Q) qm = Q - 1;
    trow[vc] = tail[qm];
    qr[vc] = rel[qm];
  }

  float sres[8];
#pragma unroll
  for (int vc = 0; vc < 8; ++vc) sres[vc] = 0.0f;

  for (int ct = 0; ct < 8; ++ct) {        // 8 column tiles of the D=128 output
    v8f acc[8] = {};                      // one accumulator per relation
    for (int k0 = 0; k0 < RGCN_D; k0 += 4) {
      const int ka = k0 + 2 * hi;
      const v2f a = *(const v2f*)(zh + ka);
      const int n = ct * 16 + lo;
#pragma unroll
      for (int r = 0; r < RGCN_R; ++r) {
        const float* Wr = relWt + (size_t)r * RGCN_MAT;
        const v2f b = *(const v2f*)(Wr + (size_t)n * RGCN_D + ka);
        acc[r] = wmma_f32_4(a, b, acc[r]);
      }
    }
    // dot with z[tail] and select the query's relation
    const int n = ct * 16 + lo;
#pragma unroll
    for (int vc = 0; vc < 8; ++vc) {
      const float ztv = z[(size_t)trow[vc] * RGCN_D + n];
      float tv = 0.0f;
#pragma unroll
      for (int r = 0; r < RGCN_R; ++r)
        tv += (qr[vc] == r) ? acc[r][vc] : 0.0f;
      sres[vc] += tv * ztv;
    }
  }

  // Reduce over the 16 lanes of each half (covers all 128 output dims)
#pragma unroll
  for (int vc = 0; vc < 8; ++vc) {
    float s = sres[vc];
    s += __shfl_xor(s, 1, 32);
    s += __shfl_xor(s, 2, 32);
    s += __shfl_xor(s, 4, 32);
    s += __shfl_xor(s, 8, 32);
    const int qm = q0 + vc + 8 * hi;
    if (lo == 0 && qm < Q) out[qm] = s;
  }
}

// ---------------------------------------------------------------------------
extern "C" void kernel_launch(void* const* d_in, const int* in_sizes, int n_in,
                              void* d_out, int out_size, void* d_ws, size_t ws_size,
                              hipStream_t stream) {
  const int*   edge_index = (const int*)d_in[0];   // [2, E]
  const int*   edge_type  = (const int*)d_in[1];   // [E]
  const int*   rel_ids    = (const int*)d_in[2];   // [Q]
  const int*   head       = (const int*)d_in[3];   // [Q]
  const int*   tail       = (const int*)d_in[4];   // [Q]
  const float* x0         = (const float*)d_in[5]; // [N, D]
  const float* W1         = (const float*)d_in[6]; // [R, D, D]
  const float* root1      = (const float*)d_in[7]; // [D, D]
  const float* b1         = (const float*)d_in[8]; // [D]
  const float* W2         = (const float*)d_in[9];
  const float* root2      = (const float*)d_in[10];
  const float* b2         = (const float*)d_in[11];
  const float* relW       = (const float*)d_in[12];
  float* out = (float*)d_out;

  const int E = in_sizes[0] / 2;
  const int Q = in_sizes[2];
  const int N = in_sizes[5] / RGCN_D;
  const int* src = edge_index;
  const int* dst = edge_index + E;

  // workspace layout
  char* ws = (char*)d_ws;
  const size_t agg_elems = (size_t)N * RGCN_KA;    // [N, R*D]
  const size_t cnt_elems = (size_t)N * RGCN_R;     // [N, R]
  float* agg   = (float*)ws;
  float* cnt   = agg + agg_elems;
  float* h     = cnt + cnt_elems;                  // [N, D]
  float* zz    = h + (size_t)N * RGCN_D;           // [N, D]
  float* W1t   = zz + (size_t)N * RGCN_D;          // [R, D, D] transposed
  float* W2t   = W1t + (size_t)RGCN_R * RGCN_MAT;
  float* r1t   = W2t + (size_t)RGCN_R * RGCN_MAT;  // [D, D] transposed
  float* r2t   = r1t + RGCN_MAT;
  float* relWt = r2t + RGCN_MAT;                   // [R, D, D] transposed

  const int blk = 256;
  const int gE      = (E + blk - 1) / blk;
  const int gNR     = ((int)cnt_elems + blk - 1) / blk;
  const int gScat   = (E * 32 + blk - 1) / blk;
  const int gLayer  = (N + 127) / 128;
  const int qtiles  = (Q + 15) / 16;
  const int gScore  = (qtiles + 7) / 8;
  const int gTrR    = (RGCN_R * RGCN_MAT + blk - 1) / blk;
  const int gTr1    = (RGCN_MAT + blk - 1) / blk;

  // weight transposes (independent of graph data)
  rgcn_transpose_kernel<<<gTrR, blk, 0, stream>>>(W1, W1t, RGCN_R);
  rgcn_transpose_kernel<<<gTrR, blk, 0, stream>>>(W2, W2t, RGCN_R);
  rgcn_transpose_kernel<<<gTr1, blk, 0, stream>>>(root1, r1t, 1);
  rgcn_transpose_kernel<<<gTr1, blk, 0, stream>>>(root2, r2t, 1);
  rgcn_transpose_kernel<<<gTrR, blk, 0, stream>>>(relW, relWt, RGCN_R);

  // counts + inverse means (shared by both layers)
  hipMemsetAsync(cnt, 0, cnt_elems * sizeof(float), stream);
  hipMemsetAsync(agg, 0, agg_elems * sizeof(float), stream);
  rgcn_count_kernel<<<gE, blk, 0, stream>>>(dst, edge_type, cnt, E);
  rgcn_inv_kernel<<<gNR, blk, 0, stream>>>(cnt, (int)cnt_elems);

  // layer 1
  rgcn_scatter_kernel<<<gScat, blk, 0, stream>>>(src, dst, edge_type, x0, agg, E);
  rgcn_layer_kernel<<<gLayer, blk, 0, stream>>>(agg, cnt, x0, W1t, r1t, b1, h, N, 1);

  // layer 2
  hipMemsetAsync(agg, 0, agg_elems * sizeof(float), stream);
  rgcn_scatter_kernel<<<gScat, blk, 0, stream>>>(src, dst, edge_type, h, agg, E);
  rgcn_layer_kernel<<<gLayer, blk, 0, stream>>>(agg, cnt, h, W2t, r2t, b2, zz, N, 0);

  // scoring
  rgcn_score_kernel<<<gScore, blk, 0, stream>>>(zz, head, tail, rel_ids, relWt, out, Q);
}